// TripletLossWithMask_41764261986536
// MI455X (gfx1250) — compile-verified
//
#include <hip/hip_runtime.h>

typedef __attribute__((ext_vector_type(2))) float v2f;
typedef __attribute__((ext_vector_type(8))) float v8f;

#define BSZ 512
#define DIM 128
#define MARGIN_F 0.2f
#define EPS_F 1e-6f

// ---------------------------------------------------------------------------
// Kernel 1: per-row squared norm and element sum (for eps cross term).
// ---------------------------------------------------------------------------
__global__ void norms_kernel(const float* __restrict__ F,
                             float* __restrict__ nrm,
                             float* __restrict__ sm) {
    int i = blockIdx.x * blockDim.x + threadIdx.x;
    if (i >= BSZ) return;
    const float4* row = (const float4*)(F + (size_t)i * DIM);
    float n = 0.f, s = 0.f;
#pragma unroll
    for (int k = 0; k < DIM / 4; ++k) {
        float4 v = row[k];
        n += v.x * v.x + v.y * v.y + v.z * v.z + v.w * v.w;
        s += v.x + v.y + v.z + v.w;
    }
    nrm[i] = n;
    sm[i] = s;
}

// ---------------------------------------------------------------------------
// Kernel 2: Gram matrix via V_WMMA_F32_16X16X4_F32 (exact fp32), fused into
// the pairwise-distance transform:
//   ||fi - fj + eps||^2 = n_i + n_j - 2*G_ij + 2*eps*(s_i - s_j) + D*eps^2
// One wave per 16x16 tile; 32x32 tiles; 8 waves per 256-thread block.
// A-fragment (16x4 f32): lane L holds row L%16, cols 2*(L/16)+{0,1}.
// B-fragment (4x16 f32): mirror layout with rows from the j-tile.
// C/D: VGPR r, lane L -> element (M = r + 8*(L/16), N = L%16).
// ---------------------------------------------------------------------------
__global__ void dist_wmma_kernel(const float* __restrict__ F,
                                 const float* __restrict__ nrm,
                                 const float* __restrict__ sm,
                                 float* __restrict__ Dmat) {
    const int wave = threadIdx.x >> 5;            // 0..7
    const int lane = threadIdx.x & 31;
    const int tile = blockIdx.x * 8 + wave;       // 0..1023
    const int ti = tile >> 5;                     // 0..31
    const int tj = tile & 31;                     // 0..31
    const int i0 = ti * 16;
    const int j0 = tj * 16;

    const int half = lane >> 4;                   // 0 or 1
    const int lrow = lane & 15;                   // 0..15

    const float* __restrict__ Ap = F + (size_t)(i0 + lrow) * DIM + 2 * half;
    const float* __restrict__ Bp = F + (size_t)(j0 + lrow) * DIM + 2 * half;

    v8f acc = {};
#pragma unroll
    for (int kk = 0; kk < DIM; kk += 4) {
        v2f a, b;
        a.x = Ap[kk];
        a.y = Ap[kk + 1];
        b.x = Bp[kk];
        b.y = Bp[kk + 1];
        // 8 args: (neg_a, A, neg_b, B, c_mod, C, reuse_a, reuse_b)
        acc = __builtin_amdgcn_wmma_f32_16x16x4_f32(
            false, a, false, b, (short)0, acc, false, false);
    }

    const int gj = j0 + lrow;                     // column index in D
    const float nj = nrm[gj];
    const float sj = sm[gj];
    const float epsterm = (float)DIM * EPS_F * EPS_F;

#pragma unroll
    for (int r = 0; r < 8; ++r) {
        const int m = r + 8 * half;
        const int gi = i0 + m;
        float t = nrm[gi] + nj - 2.0f * acc[r]
                + 2.0f * EPS_F * (sm[gi] - sj) + epsterm;
        t = fmaxf(t, 0.0f);
        Dmat[(size_t)gi * BSZ + gj] = sqrtf(t);
    }
}

// ---------------------------------------------------------------------------
// Kernel 3: per-anchor triplet sum. One 256-thread block per anchor i.
// Row D[i,:] + pos/neg flags staged in LDS; thread t owns positives
// j = t, t+256, ...; inner loop over all k (neg-masked).
// Writes per-anchor numerator and valid-count (deterministic, no atomics).
// ---------------------------------------------------------------------------
__global__ void triplet_kernel(const float* __restrict__ Dmat,
                               const int* __restrict__ mask,
                               float* __restrict__ pNum,
                               float* __restrict__ pCnt) {
    const int i = blockIdx.x;
    const int tid = threadIdx.x;                  // 0..255

    __shared__ float d[BSZ];
    __shared__ unsigned char isPos[BSZ];
    __shared__ unsigned char isNeg[BSZ];
    __shared__ float rsum[256];
    __shared__ int rpos[256];
    __shared__ int rneg[256];

    int cpos = 0, cneg = 0;
    for (int j = tid; j < BSZ; j += 256) {
        float dv = Dmat[(size_t)i * BSZ + j];
        int mv = mask[(size_t)i * BSZ + j];
        bool p = (mv != 0);
        bool ng = (!p) && (j != i);
        d[j] = dv;
        isPos[j] = (unsigned char)p;
        isNeg[j] = (unsigned char)ng;
        cpos += p ? 1 : 0;
        cneg += ng ? 1 : 0;
    }
    rpos[tid] = cpos;
    rneg[tid] = cneg;
    __syncthreads();

    float num = 0.0f;
    for (int j = tid; j < BSZ; j += 256) {
        if (isPos[j]) {
            const float dj = d[j] + MARGIN_F;
            for (int k = 0; k < BSZ; ++k) {
                float v = fmaxf(dj - d[k], 0.0f);
                num += isNeg[k] ? v : 0.0f;
            }
        }
    }
    rsum[tid] = num;
    __syncthreads();

    for (int s = 128; s > 0; s >>= 1) {
        if (tid < s) {
            rsum[tid] += rsum[tid + s];
            rpos[tid] += rpos[tid + s];
            rneg[tid] += rneg[tid + s];
        }
        __syncthreads();
    }
    if (tid == 0) {
        pNum[i] = rsum[0];
        pCnt[i] = (float)((long long)rpos[0] * (long long)rneg[0]);
    }
}

// ---------------------------------------------------------------------------
// Kernel 4: final reduction over 512 anchors, guarded division.
// ---------------------------------------------------------------------------
__global__ void finalize_kernel(const float* __restrict__ pNum,
                                const float* __restrict__ pCnt,
                                float* __restrict__ out) {
    __shared__ float sn[256];
    __shared__ float sc[256];
    const int tid = threadIdx.x;
    float n = 0.f, c = 0.f;
    for (int i = tid; i < BSZ; i += 256) {
        n += pNum[i];
        c += pCnt[i];
    }
    sn[tid] = n;
    sc[tid] = c;
    __syncthreads();
    for (int s = 128; s > 0; s >>= 1) {
        if (tid < s) {
            sn[tid] += sn[tid + s];
            sc[tid] += sc[tid + s];
        }
        __syncthreads();
    }
    if (tid == 0) {
        out[0] = (sc[0] > 0.0f) ? (sn[0] / sc[0]) : 0.0f;
    }
}

// ---------------------------------------------------------------------------
// Launch: features [512,128] f32, mask [512,512] i32 -> scalar f32.
// Workspace layout (floats): D[512*512] | nrm[512] | sm[512] | pNum[512] | pCnt[512]
// ---------------------------------------------------------------------------
extern "C" void kernel_launch(void* const* d_in, const int* in_sizes, int n_in,
                              void* d_out, int out_size, void* d_ws, size_t ws_size,
                              hipStream_t stream) {
    (void)in_sizes; (void)n_in; (void)out_size; (void)ws_size;

    const float* F    = (const float*)d_in[0];
    const int*   mask = (const int*)d_in[1];
    float*       out  = (float*)d_out;

    float* ws   = (float*)d_ws;
    float* Dmat = ws;                       // 512*512
    float* nrm  = ws + BSZ * BSZ;           // 512
    float* sm   = nrm + BSZ;                // 512
    float* pN   = sm + BSZ;                 // 512
    float* pC   = pN + BSZ;                 // 512

    norms_kernel<<<2, 256, 0, stream>>>(F, nrm, sm);
    dist_wmma_kernel<<<128, 256, 0, stream>>>(F, nrm, sm, Dmat);
    triplet_kernel<<<BSZ, 256, 0, stream>>>(Dmat, mask, pN, pC);
    finalize_kernel<<<1, 256, 0, stream>>>(pN, pC, out);
}